// ProposalGenerator_13013750907326
// MI455X (gfx1250) — compile-verified
//
#include <hip/hip_runtime.h>
#include <stdint.h>

typedef unsigned int u32x4 __attribute__((ext_vector_type(4)));
typedef int          i32x8 __attribute__((ext_vector_type(8)));
typedef int          i32x4 __attribute__((ext_vector_type(4)));

#define BATCH      16
#define CAT        80
#define NBIN       2048      // 11-bit radix per pass
#define CHUNK      5120      // floats per block; divides 80*h*w for all levels
#define NSLOT      80        // 5 levels * 16 batches
#define TOTAL_DETS 3320      // 1000+1000+1000+256+64
#define OUT_K      1000
#define SORT_N     4096

struct Cand { float s; int ind; };
struct LevelPtrs { const float* tl[5]; const float* br[5]; };

// Monotonic order-preserving key: bigger float -> bigger unsigned key.
__device__ __forceinline__ unsigned fkey(float v) {
  unsigned u = __float_as_uint(v);
  return u ^ ((u >> 31) ? 0xFFFFFFFFu : 0x80000000u);
}

__device__ __forceinline__ int level_K(int level) {
  int hw = 128 >> level; int s = hw * hw; return s < 1000 ? s : 1000;
}
__device__ __forceinline__ int level_off(int level) {
  return (level < 3) ? level * 1000 : (level == 3 ? 3000 : 3256);
}

// ---------------------------------------------------------------------------
// Stage one CHUNK of floats from global memory into LDS via the Tensor Data
// Mover. Issued by wave 0 only (TDM ignores EXEC; other waves branch around).
// D# per CDNA5 ISA #8.3/8.4:
//   group0: count=1 | lds_addr | global_addr[56:0] | type=2
//   group1: data_size=4B, tensor CHUNKx1, tile CHUNKx1, stride CHUNK
// ---------------------------------------------------------------------------
__device__ __forceinline__ void tdm_load_chunk(float* lds_dst,
                                               const float* gsrc) {
  if (threadIdx.x < 32) {
    unsigned lds_addr = (unsigned)(uintptr_t)lds_dst;  // low 32b = LDS offset
    unsigned long long ga = (unsigned long long)(uintptr_t)gsrc;
    u32x4 g0;
    g0[0] = 1u;
    g0[1] = lds_addr;
    g0[2] = (unsigned)(ga & 0xFFFFFFFFull);
    g0[3] = (unsigned)((ga >> 32) & 0x1FFFFFFull) | (2u << 30);
    i32x8 g1;
    g1[0] = (int)(2u << 16);                 // data_size=2 (4 bytes), mask=0
    g1[1] = (int)((unsigned)CHUNK << 16);    // tensor_dim0 low16 @ bits 63:48
    g1[2] = (int)(1u << 16);                 // tensor_dim0 hi=0 | tensor_dim1=1
    g1[3] = (int)((unsigned)CHUNK << 16);    // tensor_dim1 hi=0 | tile_dim0
    g1[4] = 1;                               // tile_dim1=1, tile_dim2=0
    g1[5] = CHUNK;                           // tensor_dim0_stride low32
    g1[6] = (int)((unsigned)CHUNK << 16);    // stride0 hi=0 | stride1 low16
    g1[7] = 0;
    i32x4 g2; g2[0]=0; g2[1]=0; g2[2]=0; g2[3]=0;   // group 2 unused (<=2D)
    i32x4 g3; g3[0]=0; g3[1]=0; g3[2]=0; g3[3]=0;   // group 3 unused
    i32x8 gx; gx[0]=0; gx[1]=0; gx[2]=0; gx[3]=0;
    gx[4]=0; gx[5]=0; gx[6]=0; gx[7]=0;
    __builtin_amdgcn_tensor_load_to_lds(g0, g1, g2, g3, gx, 0);
    __builtin_amdgcn_s_wait_tensorcnt(0);
  }
  __syncthreads();
}

__global__ void zero_kernel(unsigned* p, int n) {
  int i = blockIdx.x * blockDim.x + threadIdx.x;
  if (i < n) p[i] = 0u;
}

// Pass 1 histogram over top-11 key bits; heat chunk staged through TDM.
__global__ void hist1_kernel(const float* __restrict__ heat,
                             unsigned* __restrict__ hist, int level) {
  __shared__ float    sdata[CHUNK];
  __shared__ unsigned shist[NBIN];
  const int hw = 128 >> level;
  const int n  = CAT * hw * hw;
  const int b  = blockIdx.y;
  const long long base = (long long)b * n + (long long)blockIdx.x * CHUNK;

  for (int i = threadIdx.x; i < NBIN; i += blockDim.x) shist[i] = 0u;
  tdm_load_chunk(sdata, heat + base);   // includes the workgroup barrier

  for (int i = threadIdx.x; i < CHUNK; i += blockDim.x) {
    unsigned k = fkey(sdata[i]);
    atomicAdd(&shist[k >> 21], 1u);
  }
  __syncthreads();

  unsigned* gh = hist + (level * BATCH + b) * NBIN;
  for (int i = threadIdx.x; i < NBIN; i += blockDim.x) {
    unsigned c = shist[i];
    if (c) atomicAdd(&gh[i], c);
  }
}

// Pass-1 threshold: highest bin T1 s.t. count(bins >= T1) >= K.
__global__ void thresh1_kernel(const unsigned* __restrict__ hist,
                               unsigned* __restrict__ thrT1,
                               unsigned* __restrict__ thrA1) {
  __shared__ unsigned h[NBIN];
  const int slot = blockIdx.x;
  const unsigned* gh = hist + slot * NBIN;
  for (int i = threadIdx.x; i < NBIN; i += blockDim.x) h[i] = gh[i];
  __syncthreads();
  if (threadIdx.x == 0) {
    const int K = level_K(slot / BATCH);
    unsigned c = 0, above = 0; int T = 0;
    for (int bin = NBIN - 1; bin >= 0; --bin) {
      unsigned nc = c + h[bin];
      if ((int)nc >= K) { T = bin; above = c; break; }
      c = nc;
    }
    thrT1[slot] = (unsigned)T;
    thrA1[slot] = above;
  }
}

// Pass 2 histogram: refine key bits [20:10] for elements in the boundary bin.
// Heat is L2-resident on this second sweep (112 MB < 192 MB L2).
__global__ void hist2_kernel(const float* __restrict__ heat,
                             const unsigned* __restrict__ thrT1,
                             unsigned* __restrict__ hist2, int level) {
  __shared__ float    sdata[CHUNK];
  __shared__ unsigned shist[NBIN];
  const int hw = 128 >> level;
  const int n  = CAT * hw * hw;
  const int b  = blockIdx.y;
  const int slot = level * BATCH + b;
  const unsigned T1 = thrT1[slot];
  for (int i = threadIdx.x; i < NBIN; i += blockDim.x) shist[i] = 0u;
  const long long base = (long long)b * n + (long long)blockIdx.x * CHUNK;
  tdm_load_chunk(sdata, heat + base);

  for (int i = threadIdx.x; i < CHUNK; i += blockDim.x) {
    unsigned k = fkey(sdata[i]);
    if ((k >> 21) == T1) atomicAdd(&shist[(k >> 10) & 0x7FFu], 1u);
  }
  __syncthreads();
  unsigned* gh = hist2 + slot * NBIN;
  for (int i = threadIdx.x; i < NBIN; i += blockDim.x) {
    unsigned c = shist[i];
    if (c) atomicAdd(&gh[i], c);
  }
}

// Pass-2 threshold: final 22-bit key prefix threshold + counts.
__global__ void thresh2_kernel(const unsigned* __restrict__ hist2,
                               const unsigned* __restrict__ thrT1,
                               const unsigned* __restrict__ thrA1,
                               unsigned* __restrict__ thrKey,
                               unsigned* __restrict__ thrAbove,
                               unsigned* __restrict__ thrNeed) {
  __shared__ unsigned h[NBIN];
  const int slot = blockIdx.x;
  const unsigned* gh = hist2 + slot * NBIN;
  for (int i = threadIdx.x; i < NBIN; i += blockDim.x) h[i] = gh[i];
  __syncthreads();
  if (threadIdx.x == 0) {
    const int K = level_K(slot / BATCH);
    const unsigned a1 = thrA1[slot];
    const int rem = K - (int)a1;          // >= 1 by construction
    unsigned c = 0, above2 = 0; int T2 = 0;
    for (int sb = NBIN - 1; sb >= 0; --sb) {
      unsigned nc = c + h[sb];
      if ((int)nc >= rem) { T2 = sb; above2 = c; break; }
      c = nc;
    }
    thrKey[slot]   = (thrT1[slot] << 11) | (unsigned)T2;
    thrAbove[slot] = a1 + above2;
    thrNeed[slot]  = (unsigned)(rem - (int)above2);
  }
}

// Compaction: gather exactly K (score, spatial-ind) candidates per slot.
__global__ void compact_kernel(const float* __restrict__ heat,
                               const unsigned* __restrict__ thrKey,
                               const unsigned* __restrict__ thrAbove,
                               const unsigned* __restrict__ thrNeed,
                               unsigned* __restrict__ cntA,
                               unsigned* __restrict__ cntE,
                               Cand* __restrict__ cand, int level) {
  __shared__ float sdata[CHUNK];
  const int hw  = 128 >> level;
  const int hw2 = hw * hw;
  const int n   = CAT * hw2;
  const int b   = blockIdx.y;
  const int slot = level * BATCH + b;
  const unsigned TK    = thrKey[slot];
  const unsigned above = thrAbove[slot];
  const unsigned need  = thrNeed[slot];
  Cand* cslot = cand + slot * 1000;
  const long long base = (long long)b * n;
  const int e0 = blockIdx.x * CHUNK;
  tdm_load_chunk(sdata, heat + base + e0);

  for (int i = threadIdx.x; i < CHUNK; i += blockDim.x) {
    const float v = sdata[i];
    const unsigned k22 = fkey(v) >> 10;
    if (k22 < TK) continue;
    const int e = e0 + i;
    if (k22 > TK) {
      unsigned p = atomicAdd(&cntA[slot], 1u);
      cslot[p].s = v; cslot[p].ind = e % hw2;
    } else {
      unsigned p = atomicAdd(&cntE[slot], 1u);
      if (p < need) { cslot[above + p].s = v; cslot[above + p].ind = e % hw2; }
    }
  }
}

// Decode: gather tl/br regressions, box math, invalidation, write det rows.
__global__ void decode_kernel(LevelPtrs lp, const Cand* __restrict__ cand,
                              float* __restrict__ dets) {
  const int slot  = blockIdx.x;
  const int level = slot / BATCH;
  const int b     = slot % BATCH;
  const int hw  = 128 >> level;
  const int hw2 = hw * hw;
  const int K   = level_K(level);
  const float scale = (float)(8 << level);   // 8*128/hw
  const float* tl = lp.tl[level];
  const float* br = lp.br[level];
  const int off = level_off(level);
  const Cand* cslot = cand + slot * 1000;
  for (int k = threadIdx.x; k < K; k += blockDim.x) {
    const Cand c = cslot[k];
    const int ind = c.ind;
    const float ysf = (float)(ind / hw);
    const float xsf = (float)(ind % hw);
    const float tlx = tl[((long long)b * 2 + 0) * hw2 + ind];
    const float tly = tl[((long long)b * 2 + 1) * hw2 + ind];
    const float brx = br[((long long)b * 2 + 0) * hw2 + ind];
    const float bry = br[((long long)b * 2 + 1) * hw2 + ind];
    // BASE_LAYER_RANGE/8/2 = (0,4,0,4): span=4, center offset=2
    const float tlX = xsf - (4.f * tlx + 2.f);
    const float tlY = ysf - (4.f * tly + 2.f);
    const float brX = xsf + (4.f * brx + 2.f);
    const float brY = ysf + (4.f * bry + 2.f);
    float s = c.s;
    if (brX < tlX || brY < tlY) s = -1.f;
    float* row = dets + ((long long)b * TOTAL_DETS + off + k) * 7;
    row[0] = s;
    row[1] = tlX * scale; row[2] = tlY * scale;
    row[3] = brX * scale; row[4] = brY * scale;
    row[5] = 0.f;         row[6] = (float)level;
  }
}

// Final per-batch top-1000 over 3320 rows: LDS bitonic sort of 4096 pairs.
__global__ __launch_bounds__(1024)
void final_topk_kernel(const float* __restrict__ dets, float* __restrict__ out) {
  __shared__ float ss[SORT_N];
  __shared__ int   si[SORT_N];
  const int b = blockIdx.x;
  const float* db = dets + (long long)b * TOTAL_DETS * 7;
  for (int t = threadIdx.x; t < SORT_N; t += blockDim.x) {
    ss[t] = (t < TOTAL_DETS) ? db[t * 7] : -3.402823466e38f;
    si[t] = t;
  }
  for (unsigned k = 2; k <= SORT_N; k <<= 1) {
    for (unsigned j = k >> 1; j > 0; j >>= 1) {
      __syncthreads();
      for (unsigned t = threadIdx.x; t < SORT_N / 2; t += blockDim.x) {
        const unsigned i = ((t & ~(j - 1)) << 1) | (t & (j - 1));
        const unsigned p = i | j;
        const bool desc = (i & k) == 0;
        const float a = ss[i], c = ss[p];
        if ((a < c) == desc) {
          ss[i] = c; ss[p] = a;
          const int tmp = si[i]; si[i] = si[p]; si[p] = tmp;
        }
      }
    }
  }
  __syncthreads();
  if (threadIdx.x < OUT_K) {
    const int r = si[threadIdx.x];
    float* orow = out + ((long long)b * OUT_K + threadIdx.x) * 7;
    const float* drow = db + (long long)r * 7;
#pragma unroll
    for (int j = 0; j < 7; ++j) orow[j] = drow[j];
  }
}

extern "C" void kernel_launch(void* const* d_in, const int* in_sizes, int n_in,
                              void* d_out, int out_size, void* d_ws, size_t ws_size,
                              hipStream_t stream) {
  // Workspace layout (all fully rewritten or zeroed every call).
  unsigned* hist1 = (unsigned*)d_ws;                 // 80*2048 u32
  unsigned* hist2 = hist1 + NSLOT * NBIN;            // 80*2048 u32
  unsigned* cntA  = hist2 + NSLOT * NBIN;            // 80
  unsigned* cntE  = cntA  + NSLOT;                   // 80
  unsigned* thrT1 = cntE  + NSLOT;                   // 80
  unsigned* thrA1 = thrT1 + NSLOT;                   // 80
  unsigned* thrK  = thrA1 + NSLOT;                   // 80
  unsigned* thrAb = thrK  + NSLOT;                   // 80
  unsigned* thrNd = thrAb + NSLOT;                   // 80
  Cand*  cand = (Cand*)(thrNd + NSLOT);              // 80*1000 * 8B
  float* dets = (float*)(cand + NSLOT * 1000);       // 16*3320*7 f32

  const int zero_n = NSLOT * NBIN * 2 + NSLOT * 2;   // hist1+hist2+cntA+cntE
  zero_kernel<<<(zero_n + 255) / 256, 256, 0, stream>>>(hist1, zero_n);

  const int hws[5] = {128, 64, 32, 16, 8};
  for (int l = 0; l < 5; ++l) {
    const float* heat = (const float*)d_in[3 * l];
    const int n = CAT * hws[l] * hws[l];
    hist1_kernel<<<dim3(n / CHUNK, BATCH), 256, 0, stream>>>(heat, hist1, l);
  }
  thresh1_kernel<<<NSLOT, 256, 0, stream>>>(hist1, thrT1, thrA1);
  for (int l = 0; l < 5; ++l) {
    const float* heat = (const float*)d_in[3 * l];
    const int n = CAT * hws[l] * hws[l];
    hist2_kernel<<<dim3(n / CHUNK, BATCH), 256, 0, stream>>>(heat, thrT1, hist2, l);
  }
  thresh2_kernel<<<NSLOT, 256, 0, stream>>>(hist2, thrT1, thrA1, thrK, thrAb, thrNd);
  for (int l = 0; l < 5; ++l) {
    const float* heat = (const float*)d_in[3 * l];
    const int n = CAT * hws[l] * hws[l];
    compact_kernel<<<dim3(n / CHUNK, BATCH), 256, 0, stream>>>(heat, thrK, thrAb, thrNd,
                                                               cntA, cntE, cand, l);
  }
  LevelPtrs lp;
  for (int l = 0; l < 5; ++l) {
    lp.tl[l] = (const float*)d_in[3 * l + 1];
    lp.br[l] = (const float*)d_in[3 * l + 2];
  }
  decode_kernel<<<NSLOT, 256, 0, stream>>>(lp, cand, dets);
  final_topk_kernel<<<BATCH, 1024, 0, stream>>>(dets, (float*)d_out);

  (void)in_sizes; (void)n_in; (void)out_size; (void)ws_size;
}